// TransformerEmbedding_35330400977464
// MI455X (gfx1250) — compile-verified
//
#include <hip/hip_runtime.h>
#include <hip/hip_bf16.h>

typedef __attribute__((ext_vector_type(2))) float v2f;
typedef __attribute__((ext_vector_type(8))) float v8f;

#define D_EMB   256
#define D_MODEL 512
#define VOCAB   32000
#define SEQ     2048
#define NB      16
#define LN_EPS  1e-5f

#define ROWS 64      // sequence rows per block in kernel 2
#define ASTR 260     // LDS stride (floats) for A tile (16B-aligned rows, conflict-free b64 frags)
#define XSTR 516     // LDS stride (floats) for x tile

// ---------------------------------------------------------------------------
// Kernel 1: WkT[b][i][m] = sum_j Wb[m][i][j] * king_table[king_id[b]][j]
// GEMM view: C(131072 x 16) = Wb_flat(131072 x 256) x keT(256 x 16)
// One wave per 16-row M-tile, f32 WMMA 16x16x4, K-loop of 64 steps.
// ---------------------------------------------------------------------------
__global__ __launch_bounds__(256) void wk_kernel(
    const int*   __restrict__ king_id,
    const float* __restrict__ Wb,
    const float* __restrict__ king_table,
    float*       __restrict__ wkt)
{
    __shared__ float keT[D_EMB * NB];   // keT[j*16 + b]

    const int t = threadIdx.x;
    {
        const int b  = t & 15;
        const int j0 = t >> 4;                      // 0..15
        const int kb = king_id[b];
        const float* src = king_table + (size_t)kb * D_EMB;
        #pragma unroll
        for (int i = 0; i < 16; ++i) {
            const int j = j0 + i * 16;
            keT[j * 16 + b] = src[j];
        }
    }
    __syncthreads();

    const int lane = t & 31;
    const int wv   = t >> 5;          // 0..7
    const int half = lane >> 4;       // 0/1
    const int lm   = lane & 15;
    const int Rb   = blockIdx.x * 128 + wv * 16;    // tile row base (flat Wb row)

    // A frag: lane holds row (Rb+lm), K = k0 + 2*half + {0,1}  -> contiguous float2
    const float* arow = Wb + (size_t)(Rb + lm) * D_EMB + 2 * half;

    v8f acc = {0.f, 0.f, 0.f, 0.f, 0.f, 0.f, 0.f, 0.f};

    #pragma unroll 4
    for (int k0 = 0; k0 < D_EMB; k0 += 4) {
        v2f a = *(const v2f*)(arow + k0);
        const int jj = k0 + 2 * half;
        v2f bf;
        bf.x = keT[jj * 16 + lm];
        bf.y = keT[(jj + 1) * 16 + lm];
        acc = __builtin_amdgcn_wmma_f32_16x16x4_f32(
                  false, a, false, bf, (short)0, acc, false, false);
    }

    // C layout: VGPR v, lane l -> M = v + 8*half, N = lm (= batch)
    #pragma unroll
    for (int v = 0; v < 8; ++v) {
        const int R = Rb + v + 8 * half;   // flat row = m*256 + i
        const int m = R >> 8;
        const int i = R & 255;
        wkt[(size_t)lm * (D_EMB * D_MODEL) + (size_t)i * D_MODEL + m] = acc[v];
    }
}

// ---------------------------------------------------------------------------
// Kernel 2: gather tok tile -> LDS, x = tok @ WkT + bb via f32 WMMA,
// fused LayerNorm * gamma + beta.  Block = 512 thr (16 waves), 64 rows x 512 cols.
// Wave w owns cols [w*32, w*32+32): 4 M-tiles x 2 N-tiles, K = 256 in steps of 4.
// ---------------------------------------------------------------------------
__global__ __launch_bounds__(512) void emb_ln_kernel(
    const int*   __restrict__ sequence,
    const int*   __restrict__ king_id,
    const float* __restrict__ token_tables,
    const float* __restrict__ bb,
    const float* __restrict__ gamma,
    const float* __restrict__ beta,
    const float* __restrict__ wkt,
    float*       __restrict__ out)
{
    __shared__ float xs[ROWS * XSTR];           // A tile (ROWS*ASTR) then x tile
    __shared__ float red [ROWS * 8];
    __shared__ float red2[ROWS * 8];
    __shared__ float mrow[ROWS * 2];            // mean, rstd per row

    const int t  = threadIdx.x;
    const int b  = blockIdx.y;
    const int s0 = blockIdx.x * ROWS;
    const int kb = king_id[b];

    // ---- stage gathered token rows (64 x 256 f32) into LDS, stride ASTR ----
    {
        const int r = t >> 3;
        const int c = t & 7;
        const int tok = sequence[b * SEQ + s0 + r];
        const float* src = token_tables + ((size_t)kb * VOCAB + (size_t)tok) * D_EMB;
        float* dst = xs + r * ASTR;
        #pragma unroll
        for (int ii = 0; ii < 8; ++ii) {
            const int f4 = c + ii * 8;
            *(float4*)(dst + f4 * 4) = *(const float4*)(src + f4 * 4);
        }
    }
    __syncthreads();

    const int lane = t & 31;
    const int wv   = t >> 5;          // 0..15
    const int half = lane >> 4;
    const int lm   = lane & 15;
    const int col0 = wv * 32;

    const float* bbase = wkt + (size_t)b * (D_EMB * D_MODEL);

    const v8f vzero = {0.f, 0.f, 0.f, 0.f, 0.f, 0.f, 0.f, 0.f};
    v8f acc[4][2];
    #pragma unroll
    for (int mt = 0; mt < 4; ++mt)
        #pragma unroll
        for (int nt = 0; nt < 2; ++nt) acc[mt][nt] = vzero;

    #pragma unroll 2
    for (int k0 = 0; k0 < D_EMB; k0 += 4) {
        const int kk = k0 + 2 * half;
        v2f afr[4];
        #pragma unroll
        for (int mt = 0; mt < 4; ++mt)
            afr[mt] = *(const v2f*)(xs + (mt * 16 + lm) * ASTR + kk);
        v2f bfr[2];
        #pragma unroll
        for (int nt = 0; nt < 2; ++nt) {
            const int col = col0 + nt * 16 + lm;
            bfr[nt].x = bbase[(size_t)kk * D_MODEL + col];
            bfr[nt].y = bbase[(size_t)(kk + 1) * D_MODEL + col];
        }
        #pragma unroll
        for (int mt = 0; mt < 4; ++mt)
            #pragma unroll
            for (int nt = 0; nt < 2; ++nt)
                acc[mt][nt] = __builtin_amdgcn_wmma_f32_16x16x4_f32(
                    false, afr[mt], false, bfr[nt], (short)0, acc[mt][nt], false, false);
    }
    __syncthreads();   // done reading A region; xs reused for x tile

    // ---- write accumulators + bias to LDS x tile ----
    #pragma unroll
    for (int nt = 0; nt < 2; ++nt) {
        const int col  = col0 + nt * 16 + lm;
        const float bias = bb[col];
        #pragma unroll
        for (int mt = 0; mt < 4; ++mt) {
            #pragma unroll
            for (int v = 0; v < 8; ++v) {
                const int row = mt * 16 + v + 8 * half;
                xs[row * XSTR + col] = acc[mt][nt][v] + bias;
            }
        }
    }
    __syncthreads();

    // ---- LayerNorm reduce: 8 threads per row, float4 strided ----
    {
        const int r  = t >> 3;
        const int c8 = t & 7;
        float s = 0.f, ss = 0.f;
        #pragma unroll
        for (int i = 0; i < 16; ++i) {
            const int f4 = c8 + i * 8;
            float4 v = *(const float4*)(xs + r * XSTR + f4 * 4);
            s  += v.x + v.y + v.z + v.w;
            ss += v.x * v.x + v.y * v.y + v.z * v.z + v.w * v.w;
        }
        red [r * 8 + c8] = s;
        red2[r * 8 + c8] = ss;
    }
    __syncthreads();
    if ((t & 7) == 0) {
        const int r = t >> 3;
        float s = 0.f, ss = 0.f;
        #pragma unroll
        for (int c = 0; c < 8; ++c) { s += red[r * 8 + c]; ss += red2[r * 8 + c]; }
        const float mean = s * (1.f / (float)D_MODEL);
        const float var  = ss * (1.f / (float)D_MODEL) - mean * mean;
        mrow[r * 2 + 0] = mean;
        mrow[r * 2 + 1] = rsqrtf(var + LN_EPS);
    }
    __syncthreads();

    // ---- normalized, scaled, coalesced float4 stores ----
    {
        const int c4    = t & 127;      // float4 column index (0..127)
        const int rbase = t >> 7;       // 0..3
        const float4 g4 = *(const float4*)(gamma + c4 * 4);
        const float4 e4 = *(const float4*)(beta  + c4 * 4);
        #pragma unroll
        for (int i = 0; i < 16; ++i) {
            const int row  = i * 4 + rbase;
            const float mean = mrow[row * 2 + 0];
            const float rstd = mrow[row * 2 + 1];
            float4 v = *(const float4*)(xs + row * XSTR + c4 * 4);
            float4 o;
            o.x = (v.x - mean) * rstd * g4.x + e4.x;
            o.y = (v.y - mean) * rstd * g4.y + e4.y;
            o.z = (v.z - mean) * rstd * g4.z + e4.z;
            o.w = (v.w - mean) * rstd * g4.w + e4.w;
            *(float4*)(out + (size_t)(b * SEQ + s0 + row) * D_MODEL + c4 * 4) = o;
        }
    }
}

// ---------------------------------------------------------------------------
extern "C" void kernel_launch(void* const* d_in, const int* in_sizes, int n_in,
                              void* d_out, int out_size, void* d_ws, size_t ws_size,
                              hipStream_t stream) {
    const int*   sequence     = (const int*)  d_in[0];
    const int*   king_id      = (const int*)  d_in[1];
    const float* token_tables = (const float*)d_in[2];
    const float* Wb           = (const float*)d_in[3];
    const float* bbv          = (const float*)d_in[4];
    const float* king_table   = (const float*)d_in[5];
    const float* gamma        = (const float*)d_in[6];
    const float* beta         = (const float*)d_in[7];
    float*       out          = (float*)d_out;
    float*       wkt          = (float*)d_ws;   // NB * D_EMB * D_MODEL f32 = 8 MB

    // Kernel 1: per-batch transposed weight WkT[b][i][m]
    wk_kernel<<<dim3((D_MODEL * D_EMB) / 128), 256, 0, stream>>>(
        king_id, Wb, king_table, wkt);

    // Kernel 2: gather + GEMM + bias + LayerNorm
    emb_ln_kernel<<<dim3(SEQ / ROWS, NB), 512, 0, stream>>>(
        sequence, king_id, token_tables, bbv, gamma, beta, wkt, out);
}